// S4Layer_33174327395018
// MI455X (gfx1250) — compile-verified
//
#include <hip/hip_runtime.h>
#include <hip/hip_bf16.h>

typedef __attribute__((ext_vector_type(2))) float v2f;
typedef __attribute__((ext_vector_type(8))) float v8f;

#define SEQ_L 262144
#define PI_F 3.14159265358979323846f
#define PITCH 66   // even, not 0 mod 64: lane stride 2 banks, b64 -> conflict-free

// ---------------------------------------------------------------------------
// One 16x16 tile of D = F @ X, F[t][k] = exp(+2*pi*i*t*k/64) (64-pt iDFT).
// F (real/imag planes, pitch 66) is precomputed in LDS once per block.
// X is stored TRANSPOSED in LDS (lX[col*66 + k]) so that the B-operand's
// (K, K+1) pair is an adjacent, 8B-aligned ds_load_b64.
// Complex matmul via 4 real products on V_WMMA_F32_16X16X4_F32:
//   Dr = Fr@Xr - Fi@Xi   (two accumulators, VALU subtract at the end)
//   Di = Fr@Xi + Fi@Xr   (one accumulator, chained through the C operand)
// ---------------------------------------------------------------------------
__device__ __forceinline__ void dft64_tile(
    const float* __restrict__ lFr, const float* __restrict__ lFi,
    const float* __restrict__ lXr, const float* __restrict__ lXi,
    int m, int n, int lane, bool needImag, v8f& dr, v8f& di)
{
    v8f accRR = {0,0,0,0,0,0,0,0};
    v8f accII = {0,0,0,0,0,0,0,0};
    v8f accX  = {0,0,0,0,0,0,0,0};
    const int half = lane >> 4;        // lanes 16-31 hold the K+2 / M+8 halves
    const int l15  = lane & 15;
    const int trow = m * 16 + l15;     // A-matrix row (output index t)
    const int col  = n * 16 + l15;     // B/D column
    #pragma unroll
    for (int kk = 0; kk < 16; ++kk) {
        const int k0 = kk * 4 + 2 * half;       // A/B vgpr0 -> K=k0, vgpr1 -> K=k0+1
        v2f Ar = *(const v2f*)&lFr[trow * PITCH + k0];
        v2f Ai = *(const v2f*)&lFi[trow * PITCH + k0];
        v2f Br = *(const v2f*)&lXr[col  * PITCH + k0];
        v2f Bi = *(const v2f*)&lXi[col  * PITCH + k0];
        accRR = __builtin_amdgcn_wmma_f32_16x16x4_f32(false, Ar, false, Br, (short)0, accRR, false, false);
        accII = __builtin_amdgcn_wmma_f32_16x16x4_f32(false, Ai, false, Bi, (short)0, accII, false, false);
        if (needImag) {
            accX = __builtin_amdgcn_wmma_f32_16x16x4_f32(false, Ar, false, Bi, (short)0, accX, false, false);
            accX = __builtin_amdgcn_wmma_f32_16x16x4_f32(false, Ai, false, Br, (short)0, accX, false, false);
        }
    }
    dr = accRR - accII;
    di = accX;
}

// Fill the 64x64 iDFT matrix planes (16 sincos per thread, once per block).
__device__ __forceinline__ void fill_dft_matrix(float* lFr, float* lFi, int tid)
{
    const float w64 = 2.0f * PI_F / 64.0f;
    #pragma unroll
    for (int i = tid; i < 4096; i += 256) {
        int t = i >> 6, k = i & 63;
        float s, c;
        __sincosf(w64 * (float)((t * k) & 63), &s, &c);
        lFr[t * PITCH + k] = c;
        lFi[t * PITCH + k] = s;
    }
}

// ---------------------------------------------------------------------------
// Kernel 0: atRoots[l] = c * (k00 - k01*k10/(1+k11)) / L  (1/L folds the ifft
// normalization). One thread per frequency; per-mode Mnum precomputed in LDS.
// ---------------------------------------------------------------------------
__global__ __launch_bounds__(256) void s4_atroots(
    const float* __restrict__ Lam, const float* __restrict__ P,
    const float* __restrict__ B,   const float* __restrict__ C,
    const float* __restrict__ logdt, float2* __restrict__ Z)
{
    __shared__ float sM[64 * 8];
    __shared__ float sLam[128];
    const int tid = threadIdx.x;
    if (tid < 64) {
        float lr = Lam[2*tid], li = Lam[2*tid+1];
        sLam[2*tid] = lr; sLam[2*tid+1] = li;
        float pr = P[2*tid], pi = P[2*tid+1];
        float br = B[2*tid], bi = B[2*tid+1];
        float cr = C[2*tid], ci = C[2*tid+1];
        // cc = conj(C), cq = conj(P); Mnum = {cc*B, cc*P, cq*B, cq*P}
        sM[tid*8+0] = cr*br + ci*bi;  sM[tid*8+1] = cr*bi - ci*br;
        sM[tid*8+2] = cr*pr + ci*pi;  sM[tid*8+3] = cr*pi - ci*pr;
        sM[tid*8+4] = pr*br + pi*bi;  sM[tid*8+5] = pr*bi - pi*br;
        sM[tid*8+6] = pr*pr + pi*pi;  sM[tid*8+7] = 0.0f;   // conj(P)*P = |P|^2
    }
    __syncthreads();

    const int l = blockIdx.x * 256 + tid;
    const float twoOverDelta = 2.0f / __expf(logdt[0]);
    float s, c;
    __sincosf(-2.0f * PI_F * (float)l / (float)SEQ_L, &s, &c);   // Omega = c + i*s
    const float d1r = 1.0f + c, d1i = s;                          // 1 + Omega
    const float inv = 1.0f / (d1r*d1r + d1i*d1i);
    const float n1r = 1.0f - c, n1i = -s;                         // 1 - Omega
    const float gr = twoOverDelta * (n1r*d1r + n1i*d1i) * inv;
    const float gi = twoOverDelta * (n1i*d1r - n1r*d1i) * inv;
    const float cfr =  2.0f * d1r * inv;                          // c = 2/(1+Omega)
    const float cfi = -2.0f * d1i * inv;

    float k00r=0,k00i=0,k01r=0,k01i=0,k10r=0,k10i=0,k11r=0,k11i=0;
    #pragma unroll 8
    for (int n = 0; n < 64; ++n) {
        float er = gr - sLam[2*n], ei = gi - sLam[2*n+1];
        float rinv = 1.0f / (er*er + ei*ei);
        float rr = er * rinv, ri = -ei * rinv;                    // r = 1/(g-Lam)
        const float* Mn = &sM[n*8];
        k00r += rr*Mn[0] - ri*Mn[1]; k00i += rr*Mn[1] + ri*Mn[0];
        k01r += rr*Mn[2] - ri*Mn[3]; k01i += rr*Mn[3] + ri*Mn[2];
        k10r += rr*Mn[4] - ri*Mn[5]; k10i += rr*Mn[5] + ri*Mn[4];
        k11r += rr*Mn[6] - ri*Mn[7]; k11i += rr*Mn[7] + ri*Mn[6];
    }
    const float tr = k01r*k10r - k01i*k10i, ti = k01r*k10i + k01i*k10r;
    const float dnr = 1.0f + k11r, dni = k11i;
    const float dinv = 1.0f / (dnr*dnr + dni*dni);
    const float qr = (tr*dnr + ti*dni) * dinv;
    const float qi = (ti*dnr - tr*dni) * dinv;
    const float ar = k00r - qr, ai = k00i - qi;
    const float xr = cfr*ar - cfi*ai, xi = cfr*ai + cfi*ar;
    const float scale = 1.0f / (float)SEQ_L;
    Z[l] = make_float2(xr * scale, xi * scale);
}

// ---------------------------------------------------------------------------
// FFT pass 1: DFT-64 over digit l2 (stride 4096). Block wg owns l1 = wg,
// columns l0 = 0..63. Twiddle W_4096^{wg*t2} depends only on t2 -> 64 values,
// precomputed once per block in LDS.
// ---------------------------------------------------------------------------
__global__ __launch_bounds__(256) void s4_fft_stage1(float2* __restrict__ Z)
{
    __shared__ __align__(16) float lFr[64 * PITCH];
    __shared__ __align__(16) float lFi[64 * PITCH];
    __shared__ __align__(16) float lXr[64 * PITCH];   // transposed: [col][k]
    __shared__ __align__(16) float lXi[64 * PITCH];
    __shared__ float lTr[64], lTi[64];
    const int tid = threadIdx.x;
    const int wg = blockIdx.x;
    fill_dft_matrix(lFr, lFi, tid);
    if (tid < 64) {
        float s, c;
        __sincosf((2.0f * PI_F / 4096.0f) * (float)((wg * tid) & 4095), &s, &c);
        lTr[tid] = c; lTi[tid] = s;
    }
    for (int i = tid; i < 4096; i += 256) {
        int r = i >> 6, cc = i & 63;                 // r = l2 (K dim), cc = l0 (col)
        float2 v = Z[r * 4096 + wg * 64 + cc];       // coalesced in cc
        lXr[cc * PITCH + r] = v.x;                   // transposed, stride 66
        lXi[cc * PITCH + r] = v.y;
    }
    __syncthreads();
    const int lane = tid & 31, wave = tid >> 5;
    const int half = lane >> 4, l15 = lane & 15;
    for (int tt = 0; tt < 2; ++tt) {
        const int tile = wave * 2 + tt;
        const int m = tile >> 2, n = tile & 3;
        v8f dr, di;
        dft64_tile(lFr, lFi, lXr, lXi, m, n, lane, true, dr, di);
        #pragma unroll
        for (int v = 0; v < 8; ++v) {
            const int t2 = m * 16 + v + 8 * half;
            float c = lTr[t2], s = lTi[t2];          // lane-broadcast read
            float outr = dr[v] * c - di[v] * s;
            float outi = dr[v] * s + di[v] * c;
            Z[t2 * 4096 + wg * 64 + n * 16 + l15] = make_float2(outr, outi);
        }
    }
}

// ---------------------------------------------------------------------------
// FFT pass 2: DFT-64 over digit l1. Block wg owns t2 = wg. Twiddle
// W_L^{l0*(64*t1 + t2)} is genuinely per-element (4096 distinct values) ->
// computed in-register.
// ---------------------------------------------------------------------------
__global__ __launch_bounds__(256) void s4_fft_stage2(float2* __restrict__ Z)
{
    __shared__ __align__(16) float lFr[64 * PITCH];
    __shared__ __align__(16) float lFi[64 * PITCH];
    __shared__ __align__(16) float lXr[64 * PITCH];   // transposed: [col][k]
    __shared__ __align__(16) float lXi[64 * PITCH];
    const int tid = threadIdx.x;
    const int wg = blockIdx.x;
    fill_dft_matrix(lFr, lFi, tid);
    for (int i = tid; i < 4096; i += 256) {
        int r = i >> 6, cc = i & 63;                 // r = l1 (K dim), cc = l0 (col)
        float2 v = Z[wg * 4096 + r * 64 + cc];       // coalesced in cc
        lXr[cc * PITCH + r] = v.x;
        lXi[cc * PITCH + r] = v.y;
    }
    __syncthreads();
    const int lane = tid & 31, wave = tid >> 5;
    const int half = lane >> 4, l15 = lane & 15;
    const float wL = 2.0f * PI_F / (float)SEQ_L;
    for (int tt = 0; tt < 2; ++tt) {
        const int tile = wave * 2 + tt;
        const int m = tile >> 2, n = tile & 3;
        const int l0 = n * 16 + l15;                 // column = l0
        v8f dr, di;
        dft64_tile(lFr, lFi, lXr, lXi, m, n, lane, true, dr, di);
        #pragma unroll
        for (int v = 0; v < 8; ++v) {
            const int t1 = m * 16 + v + 8 * half;
            const int e = (l0 * (64 * t1 + wg)) & (SEQ_L - 1);
            float s, c;
            __sincosf(wL * (float)e, &s, &c);
            float outr = dr[v] * c - di[v] * s;
            float outi = dr[v] * s + di[v] * c;
            Z[wg * 4096 + t1 * 64 + l0] = make_float2(outr, outi);
        }
    }
}

// ---------------------------------------------------------------------------
// FFT pass 3: DFT-64 over contiguous digit l0; only the real part is needed
// (reference takes ifft(...).real), so half the WMMA work is skipped.
// Block wg owns t1 = wg, columns j = t2 = 0..63. Output is digit-reversed:
// out[t0*4096 + t1*64 + t2] -> contiguous in t2 across lanes (coalesced).
// ---------------------------------------------------------------------------
__global__ __launch_bounds__(256) void s4_fft_stage3(
    const float2* __restrict__ Z, float* __restrict__ out)
{
    __shared__ __align__(16) float lFr[64 * PITCH];
    __shared__ __align__(16) float lFi[64 * PITCH];
    __shared__ __align__(16) float lXr[64 * PITCH];   // [col=j][k=l0]
    __shared__ __align__(16) float lXi[64 * PITCH];
    const int tid = threadIdx.x;
    const int wg = blockIdx.x;
    fill_dft_matrix(lFr, lFi, tid);
    for (int i = tid; i < 4096; i += 256) {
        int j = i >> 6, l0 = i & 63;                 // j = t2 (col), l0 = K dim
        float2 v = Z[j * 4096 + wg * 64 + l0];       // coalesced in l0
        lXr[j * PITCH + l0] = v.x;                   // consecutive -> conflict-free
        lXi[j * PITCH + l0] = v.y;
    }
    __syncthreads();
    const int lane = tid & 31, wave = tid >> 5;
    const int half = lane >> 4, l15 = lane & 15;
    for (int tt = 0; tt < 2; ++tt) {
        const int tile = wave * 2 + tt;
        const int m = tile >> 2, n = tile & 3;
        v8f dr, di;
        dft64_tile(lFr, lFi, lXr, lXi, m, n, lane, false, dr, di);  // real only
        #pragma unroll
        for (int v = 0; v < 8; ++v) {
            const int t0 = m * 16 + v + 8 * half;
            out[t0 * 4096 + wg * 64 + n * 16 + l15] = dr[v];
        }
    }
}

extern "C" void kernel_launch(void* const* d_in, const int* in_sizes, int n_in,
                              void* d_out, int out_size, void* d_ws, size_t ws_size,
                              hipStream_t stream) {
    const float* Lam   = (const float*)d_in[0];
    const float* P     = (const float*)d_in[1];
    const float* B     = (const float*)d_in[2];
    const float* C     = (const float*)d_in[3];
    const float* logdt = (const float*)d_in[4];
    float2* Z = (float2*)d_ws;                 // 262144 complex = 2 MB scratch

    s4_atroots   <<<SEQ_L / 256, 256, 0, stream>>>(Lam, P, B, C, logdt, Z);
    s4_fft_stage1<<<64, 256, 0, stream>>>(Z);
    s4_fft_stage2<<<64, 256, 0, stream>>>(Z);
    s4_fft_stage3<<<64, 256, 0, stream>>>(Z, (float*)d_out);
}